// _FFTConv_21268678050000
// MI455X (gfx1250) — compile-verified
//
#include <hip/hip_runtime.h>
#include <hip/hip_bf16.h>
#include <math.h>

// Problem constants
#define BATCH   32
#define CIN     64
#define COUT    128
#define HH      128
#define WW      128
#define WF      65            // W/2+1
#define NBIN    (HH*WF)       // 8320
#define HO      120
#define WO      120
#define NIMG_IN  (BATCH*CIN)   // 2048
#define NIMG_OUT (BATCH*COUT)  // 4096
#define TWO_PI  6.283185307179586476925f

typedef __attribute__((ext_vector_type(2))) float v2f;
typedef __attribute__((ext_vector_type(8))) float v8f;

// ---------------------------------------------------------------------------
// WMMA wrapper: D = A(16x4 f32) x B(4x16 f32) + C(16x16 f32)
// ---------------------------------------------------------------------------
__device__ __forceinline__ v8f wmma_f32_16x16x4(v2f a, v2f b, v8f c) {
#if defined(__gfx1250__) && __has_builtin(__builtin_amdgcn_wmma_f32_16x16x4_f32)
    return __builtin_amdgcn_wmma_f32_16x16x4_f32(false, a, false, b, (short)0, c,
                                                 false, false);
#else
    return c;
#endif
}

// ---------------------------------------------------------------------------
// In-LDS 128-point complex FFT over 32 "lines" (cooperative, 256 threads).
// Element (line, idx) lives at re[line*LS + idx], im[line*LS + idx].
// Natural-order in, natural-order out (bit-reverse + 7 DIT stages).
// Twiddle table: twr/twi[t] = exp(sgn * i * 2*pi * t / 128), t=0..63.
// ---------------------------------------------------------------------------
__device__ __forceinline__ void fft128_lds(float* re, float* im,
                                           const float* twr, const float* twi,
                                           int tid, int LS) {
    for (int e = tid; e < 32 * 128; e += 256) {
        int line = e >> 7;
        int p    = e & 127;
        int rp   = (int)(__brev((unsigned)p) >> 25);   // 7-bit reverse
        if (rp > p) {
            int a = line * LS + p, b = line * LS + rp;
            float tr = re[a]; re[a] = re[b]; re[b] = tr;
            float ti = im[a]; im[a] = im[b]; im[b] = ti;
        }
    }
    __syncthreads();
    for (int s = 0; s < 7; ++s) {
        int m = 1 << s;
        for (int e = tid; e < 32 * 64; e += 256) {
            int line = e >> 6;
            int bf   = e & 63;
            int j    = bf & (m - 1);
            int i0   = ((bf >> s) << (s + 1)) + j;
            int i1   = i0 + m;
            int t    = j << (6 - s);
            float wr = twr[t], wi = twi[t];
            int a0 = line * LS + i0, a1 = line * LS + i1;
            float xr = re[a1], xi = im[a1];
            float tr = wr * xr - wi * xi;
            float ti = wr * xi + wi * xr;
            float ur = re[a0], ui = im[a0];
            re[a0] = ur + tr; im[a0] = ui + ti;
            re[a1] = ur - tr; im[a1] = ui - ti;
        }
        __syncthreads();
    }
}

// ---------------------------------------------------------------------------
// K1: forward row FFT. 32 rows per block (rows = (img,h) pairs).
// ---------------------------------------------------------------------------
__global__ __launch_bounds__(256) void k_rowfft_fwd(const float* __restrict__ sig,
                                                    float* __restrict__ T1r,
                                                    float* __restrict__ T1i) {
    __shared__ float re[32 * 128], im[32 * 128];
    __shared__ float twr[64], twi[64];
    int tid = threadIdx.x;
    size_t rowbase = (size_t)blockIdx.x * 32;

    for (int e = tid; e < 32 * 128; e += 256) {
        re[e] = sig[rowbase * 128 + (size_t)e];
        im[e] = 0.0f;
    }
    if (tid < 64) {
        float ang = -TWO_PI * (float)tid / 128.0f;
        twr[tid] = cosf(ang);
        twi[tid] = sinf(ang);
    }
    __syncthreads();

    fft128_lds(re, im, twr, twi, tid, 128);

    for (int e = tid; e < 32 * WF; e += 256) {
        int line = e / WF, kw = e % WF;
        size_t g   = rowbase + (size_t)line;
        size_t img = g >> 7;
        int    h   = (int)(g & 127);
        size_t o   = img * NBIN + (size_t)h * WF + (size_t)kw;
        T1r[o] = re[line * 128 + kw];
        T1i[o] = im[line * 128 + kw];
    }
}

// ---------------------------------------------------------------------------
// K2/K4: column FFT along H (complex), 32 kw-columns per block, 3 blocks/image.
// ---------------------------------------------------------------------------
__global__ __launch_bounds__(256) void k_colfft(const float* __restrict__ inr,
                                                const float* __restrict__ ini,
                                                float* __restrict__ outr,
                                                float* __restrict__ outi,
                                                float sgn) {
    __shared__ float re[32 * 129], im[32 * 129];   // stride 129: conflict-free
    __shared__ float twr[64], twi[64];
    int tid = threadIdx.x;
    int img = blockIdx.x / 3;
    int c0  = (blockIdx.x % 3) * 32;
    size_t base = (size_t)img * NBIN;

    for (int e = tid; e < 32 * 128; e += 256) {
        int h = e >> 5, col = e & 31;
        int kw = c0 + col;
        float vr = 0.0f, vi = 0.0f;
        if (kw < WF) {
            size_t idx = base + (size_t)h * WF + (size_t)kw;
            vr = inr[idx]; vi = ini[idx];
        }
        re[col * 129 + h] = vr;
        im[col * 129 + h] = vi;
    }
    if (tid < 64) {
        float ang = sgn * TWO_PI * (float)tid / 128.0f;
        twr[tid] = cosf(ang);
        twi[tid] = sinf(ang);
    }
    __syncthreads();

    fft128_lds(re, im, twr, twi, tid, 129);

    for (int e = tid; e < 32 * 128; e += 256) {
        int kh = e >> 5, col = e & 31;
        int kw = c0 + col;
        if (kw < WF) {
            size_t o = base + (size_t)kh * WF + (size_t)kw;
            outr[o] = re[col * 129 + kh];
            outi[o] = im[col * 129 + kh];
        }
    }
}

// ---------------------------------------------------------------------------
// K3: per-frequency-bin complex GEMM on WMMA f32 16x16x4.
//   out[b,o,bin] = sum_i sig[b,i,bin] * K[o,i,bin]   (complex)
// Grid: 520 blocks = 260 bin-groups x 2 Cout-halves (concurrency for latency
// overlap). Each block: 32 consecutive bins, software-pipelined: next bin's
// A (32x64) + B-half (64x64) staged through registers into double-buffered
// LDS (96 KB dynamic) while the current bin's 64 WMMAs issue. One barrier/bin.
// 8 waves, each owns one 16x16 complex output tile per bin.
// ---------------------------------------------------------------------------
__global__ __launch_bounds__(256) void k_bingemm(const float* __restrict__ Sr,
                                                 const float* __restrict__ Si,
                                                 const float* __restrict__ Wr,
                                                 const float* __restrict__ Wi,
                                                 float* __restrict__ Or_,
                                                 float* __restrict__ Oi_) {
    extern __shared__ float smem[];
    float* lAr = smem;                 // 2 x 2048
    float* lAi = lAr + 2 * 2048;       // 2 x 2048
    float* lBr = lAi + 2 * 2048;       // 2 x 4096
    float* lBi = lBr + 2 * 4096;       // 2 x 4096

    int tid  = threadIdx.x;
    int lane = tid & 31;
    int wv   = tid >> 5;               // 0..7
    int mt   = wv >> 2;                // 0..1  (b-tile)
    int nt   = wv & 3;                 // 0..3  (o-tile within half)
    int hl   = lane >> 4;              // lane half (K split / M split)
    int l15  = lane & 15;
    int mrow = mt * 16 + l15;          // A row (b) for this lane

    int grp  = blockIdx.x >> 1;        // bin group 0..259
    int hf   = blockIdx.x & 1;         // Cout half
    int o0   = hf * 64;
    int bin0 = grp * 32;

    float aR[8], aI[8], bR[16], bI[16];

    // ---- stage bin0 into registers, then LDS buffer 0 ----
#pragma unroll
    for (int j = 0; j < 8; ++j) {
        int e = tid + j * 256;
        aR[j] = Sr[(size_t)e * NBIN + (size_t)bin0];
        aI[j] = Si[(size_t)e * NBIN + (size_t)bin0];
    }
#pragma unroll
    for (int j = 0; j < 16; ++j) {
        int e = tid + j * 256;
        int i = e >> 6, oo = e & 63;
        size_t widx = ((size_t)(o0 + oo) * CIN + (size_t)i) * NBIN + (size_t)bin0;
        bR[j] = Wr[widx];
        bI[j] = Wi[widx];
    }
#pragma unroll
    for (int j = 0; j < 8; ++j) {
        int e = tid + j * 256;
        lAr[e] = aR[j]; lAi[e] = aI[j];
    }
#pragma unroll
    for (int j = 0; j < 16; ++j) {
        int e = tid + j * 256;
        lBr[e] = bR[j]; lBi[e] = bI[j];
    }
    __syncthreads();

    for (int bi = 0; bi < 32; ++bi) {
        int bin = bin0 + bi;
        int p   = bi & 1;
        const float* pAr = lAr + p * 2048;
        const float* pAi = lAi + p * 2048;
        const float* pBr = lBr + p * 4096;
        const float* pBi = lBi + p * 4096;

        // ---- issue next bin's global loads (hide latency under WMMAs) ----
        if (bi < 31) {
            int nbin = bin + 1;
#pragma unroll
            for (int j = 0; j < 8; ++j) {
                int e = tid + j * 256;
                aR[j] = Sr[(size_t)e * NBIN + (size_t)nbin];
                aI[j] = Si[(size_t)e * NBIN + (size_t)nbin];
            }
#pragma unroll
            for (int j = 0; j < 16; ++j) {
                int e = tid + j * 256;
                int i = e >> 6, oo = e & 63;
                size_t widx = ((size_t)(o0 + oo) * CIN + (size_t)i) * NBIN + (size_t)nbin;
                bR[j] = Wr[widx];
                bI[j] = Wi[widx];
            }
        }

        // ---- compute this bin's 16x16 complex tile (one per wave) ----
        v8f a_rr = {0,0,0,0,0,0,0,0};
        v8f a_ii = {0,0,0,0,0,0,0,0};
        v8f a_ri = {0,0,0,0,0,0,0,0};
        v8f a_ir = {0,0,0,0,0,0,0,0};
        int ncol = nt * 16 + l15;
#pragma unroll
        for (int k0 = 0; k0 < 64; k0 += 4) {
            int ka = k0 + 2 * hl;
            v2f ar, ai, br, bb;
            ar[0] = pAr[mrow * 64 + ka];
            ar[1] = pAr[mrow * 64 + ka + 1];
            ai[0] = pAi[mrow * 64 + ka];
            ai[1] = pAi[mrow * 64 + ka + 1];
            br[0] = pBr[ka * 64 + ncol];
            br[1] = pBr[(ka + 1) * 64 + ncol];
            bb[0] = pBi[ka * 64 + ncol];
            bb[1] = pBi[(ka + 1) * 64 + ncol];
            a_rr = wmma_f32_16x16x4(ar, br, a_rr);
            a_ii = wmma_f32_16x16x4(ai, bb, a_ii);
            a_ri = wmma_f32_16x16x4(ar, bb, a_ri);
            a_ir = wmma_f32_16x16x4(ai, br, a_ir);
        }
        // C/D layout: lanes 0-15 -> M=r, lanes 16-31 -> M=r+8; N = lane&15.
#pragma unroll
        for (int r = 0; r < 8; ++r) {
            int b = mt * 16 + hl * 8 + r;
            int o = o0 + nt * 16 + l15;
            size_t idx = ((size_t)(b * COUT + o)) * NBIN + (size_t)bin;
            Or_[idx] = a_rr[r] - a_ii[r];
            Oi_[idx] = a_ri[r] + a_ir[r];
        }

        // ---- commit staged registers to the other LDS buffer ----
        if (bi < 31) {
            int q = 1 - p;
            float* qAr = lAr + q * 2048;
            float* qAi = lAi + q * 2048;
            float* qBr = lBr + q * 4096;
            float* qBi = lBi + q * 4096;
#pragma unroll
            for (int j = 0; j < 8; ++j) {
                int e = tid + j * 256;
                qAr[e] = aR[j]; qAi[e] = aI[j];
            }
#pragma unroll
            for (int j = 0; j < 16; ++j) {
                int e = tid + j * 256;
                qBr[e] = bR[j]; qBi[e] = bI[j];
            }
        }
        __syncthreads();
    }
}

// ---------------------------------------------------------------------------
// K5: inverse row FFT + Hermitian reconstruction + scale + crop + bias.
// ---------------------------------------------------------------------------
__global__ __launch_bounds__(256) void k_rowifft(const float* __restrict__ T2r,
                                                 const float* __restrict__ T2i,
                                                 const float* __restrict__ bias,
                                                 float* __restrict__ out) {
    __shared__ float re[32 * 128], im[32 * 128];
    __shared__ float twr[64], twi[64];
    int tid = threadIdx.x;
    size_t rowbase = (size_t)blockIdx.x * 32;

    for (int e = tid; e < 32 * WF; e += 256) {
        int line = e / WF, kw = e % WF;
        size_t g  = rowbase + (size_t)line;
        size_t bo = g / HO;
        int    h  = (int)(g % HO);
        size_t idx = bo * NBIN + (size_t)h * WF + (size_t)kw;
        re[line * 128 + kw] = T2r[idx];
        im[line * 128 + kw] = T2i[idx];
    }
    if (tid < 64) {
        float ang = TWO_PI * (float)tid / 128.0f;   // inverse: conj twiddles
        twr[tid] = cosf(ang);
        twi[tid] = sinf(ang);
    }
    __syncthreads();
    for (int e = tid; e < 32 * 63; e += 256) {
        int line = e / 63, j = e % 63;
        int w = 65 + j, src = 128 - w;
        re[line * 128 + w] =  re[line * 128 + src];
        im[line * 128 + w] = -im[line * 128 + src];
    }
    __syncthreads();

    fft128_lds(re, im, twr, twi, tid, 128);

    const float scale = 1.0f / 16384.0f;   // 1/(128*128)
    for (int e = tid; e < 32 * WO; e += 256) {
        int line = e / WO, w = e % WO;
        size_t g  = rowbase + (size_t)line;
        size_t bo = g / HO;
        int    h  = (int)(g % HO);
        int    o  = (int)(bo & (COUT - 1));
        out[(bo * HO + (size_t)h) * WO + (size_t)w] =
            re[line * 128 + w] * scale + bias[o];
    }
}

// ---------------------------------------------------------------------------
// Host-side launcher
// ---------------------------------------------------------------------------
extern "C" void kernel_launch(void* const* d_in, const int* in_sizes, int n_in,
                              void* d_out, int out_size, void* d_ws, size_t ws_size,
                              hipStream_t stream) {
    (void)in_sizes; (void)n_in; (void)out_size; (void)ws_size;
    const float* signal = (const float*)d_in[0];
    const float* wreal  = (const float*)d_in[1];
    const float* wimag  = (const float*)d_in[2];
    const float* bias   = (const float*)d_in[3];
    float* out = (float*)d_out;

    float* ws = (float*)d_ws;
    const size_t nIn  = (size_t)NIMG_IN  * NBIN;   // 2048*8320
    const size_t nOut = (size_t)NIMG_OUT * NBIN;   // 4096*8320
    float* T1r = ws;
    float* T1i = T1r + nIn;
    float* Sr  = T1i + nIn;
    float* Si  = Sr  + nIn;
    float* Or_ = Si  + nIn;
    float* Oi_ = Or_ + nOut;
    float* T2r = Oi_ + nOut;
    float* T2i = T2r + nOut;

    // 1) forward row FFT: 262144 rows / 32 per block
    k_rowfft_fwd<<<(NIMG_IN * HH) / 32, 256, 0, stream>>>(signal, T1r, T1i);
    // 2) forward column FFT: 2048 images * 3 column chunks
    k_colfft<<<NIMG_IN * 3, 256, 0, stream>>>(T1r, T1i, Sr, Si, -1.0f);
    // 3) per-bin complex channel GEMM (WMMA): 260 bin-groups x 2 Cout-halves,
    //    96 KB dynamic LDS (double-buffered A and B-half)
    k_bingemm<<<(NBIN / 32) * 2, 256, 96 * 1024, stream>>>(Sr, Si, wreal, wimag,
                                                           Or_, Oi_);
    // 4) inverse column FFT: 4096 images * 3 column chunks
    k_colfft<<<NIMG_OUT * 3, 256, 0, stream>>>(Or_, Oi_, T2r, T2i, 1.0f);
    // 5) inverse row FFT + crop + bias: 4096*120 rows / 32 per block
    k_rowifft<<<(NIMG_OUT * HO) / 32, 256, 0, stream>>>(T2r, T2i, bias, out);
}